// PFNLayerV2_1872605741354
// MI455X (gfx1250) — compile-verified
//
#include <hip/hip_runtime.h>

typedef __attribute__((ext_vector_type(2))) float v2f;
typedef __attribute__((ext_vector_type(8))) float v8f;

#define C_IN   10
#define NMOM   55          // upper-triangular 10x10
#define WS_SUM   0         // 10 floats
#define WS_MOM   16        // 55 floats
#define WS_SCALE 128       // 32 floats
#define WS_BIAS  160       // 32 floats
#define WS_XMAX  256       // num_segments*32 floats
#define BN_EPS 1e-3f

// ---------------- init: zero accumulators + segment-max table ----------------
__global__ void pfn_init_ws(float* __restrict__ ws, const int* __restrict__ nsegp) {
    long long total = (long long)WS_XMAX + (long long)(*nsegp) * 32;
    long long stride = (long long)gridDim.x * blockDim.x;
    for (long long i = (long long)blockIdx.x * blockDim.x + threadIdx.x; i < total; i += stride)
        ws[i] = 0.0f;
}

// ---------------- pass 1: input moments (sum_i a_i, sum_i a_i a_j) ----------------
__global__ void __launch_bounds__(256) pfn_moments(const float* __restrict__ in,
                                                   float* __restrict__ ws, int N) {
    __shared__ float s[C_IN + NMOM];
    if (threadIdx.x < C_IN + NMOM) s[threadIdx.x] = 0.0f;
    __syncthreads();

    float sum[C_IN];
    float mom[NMOM];
#pragma unroll
    for (int i = 0; i < C_IN; i++) sum[i] = 0.0f;
#pragma unroll
    for (int i = 0; i < NMOM; i++) mom[i] = 0.0f;

    int stride = gridDim.x * blockDim.x;
    for (int r = blockIdx.x * blockDim.x + threadIdx.x; r < N; r += stride) {
        float a[C_IN];
        const float2* p = (const float2*)(in + (long long)r * C_IN);  // 40B rows -> 8B aligned
#pragma unroll
        for (int i = 0; i < 5; i++) { float2 v = p[i]; a[2 * i] = v.x; a[2 * i + 1] = v.y; }
#pragma unroll
        for (int i = 0; i < C_IN; i++) sum[i] += a[i];
        int t = 0;
#pragma unroll
        for (int i = 0; i < C_IN; i++)
#pragma unroll
            for (int j = i; j < C_IN; j++) mom[t++] += a[i] * a[j];
    }
#pragma unroll
    for (int i = 0; i < C_IN; i++) atomicAdd(&s[i], sum[i]);
#pragma unroll
    for (int i = 0; i < NMOM; i++) atomicAdd(&s[C_IN + i], mom[i]);
    __syncthreads();
    if (threadIdx.x < C_IN)
        atomicAdd(&ws[WS_SUM + threadIdx.x], s[threadIdx.x]);
    else if (threadIdx.x < C_IN + NMOM)
        atomicAdd(&ws[WS_MOM + (threadIdx.x - C_IN)], s[threadIdx.x]);
}

// ---------------- pass 2: fold moments + gamma/beta into scale/bias ----------------
__global__ void pfn_finalize(const float* __restrict__ W, const float* __restrict__ gamma,
                             const float* __restrict__ beta, float* __restrict__ ws, float invN) {
    int c = threadIdx.x;
    if (c >= 32) return;
    float w[C_IN];
#pragma unroll
    for (int i = 0; i < C_IN; i++) w[i] = W[c * C_IN + i];
    float mean = 0.0f;
#pragma unroll
    for (int i = 0; i < C_IN; i++) mean += w[i] * ws[WS_SUM + i];
    mean *= invN;
    float ex2 = 0.0f;
    int t = 0;
#pragma unroll
    for (int i = 0; i < C_IN; i++)
#pragma unroll
        for (int j = i; j < C_IN; j++) {
            float m = ws[WS_MOM + t++];
            ex2 += ((i == j) ? 1.0f : 2.0f) * w[i] * w[j] * m;
        }
    ex2 *= invN;
    float var = ex2 - mean * mean;
    float sc = gamma[c] * rsqrtf(var + BN_EPS);
    ws[WS_SCALE + c] = sc;
    ws[WS_BIAS + c] = beta[c] - mean * sc;
}

// ---------------- pass 3: WMMA linear + BN affine + ReLU + store + scatter-max ----------------
// Per wave: 16 points x 32 channels via V_WMMA_F32_16X16X4_F32, K=10 padded to 12.
// A 16x4 f32: lane(M)=0..15, {vgpr v, lane-half h} -> K = 4*ks + v + 2h.
// B  4x16 f32: lane(N)=0..15, same K mapping.
// C/D 16x16 f32: vgpr r -> M = r + 8h, N = lane&15.
// k-step 2 wants k = 8+2h: only h==0 is real data, so load at k=8 for ALL lanes
// (always in-bounds) and cndmask-zero the h==1 lanes -- no divergent branches.
__global__ void __launch_bounds__(256) pfn_main(const float* __restrict__ in,
                                                const float* __restrict__ W,
                                                const int* __restrict__ unq,
                                                float* __restrict__ ws,
                                                float* __restrict__ out, int N) {
    const int lane = threadIdx.x & 31;
    const int wv   = threadIdx.x >> 5;
    const int half = lane >> 4;
    const int mr   = lane & 15;
    const int kh   = 2 * half;
    const int row0 = (blockIdx.x * 8 + wv) * 16;          // < 2^31/64, int-safe
    const bool full = (row0 + 16 <= N);

    // ---- B fragments: all loads unconditional, tail lanes zeroed by select
    v2f b0[3], b1[3];
    {
        const float* w0 = W + mr * C_IN;
        const float* w1 = W + (16 + mr) * C_IN;
        float2 v00 = *(const float2*)(w0 + kh);
        float2 v01 = *(const float2*)(w0 + 4 + kh);
        float2 v02 = *(const float2*)(w0 + 8);            // k=8 valid for all lanes
        float2 v10 = *(const float2*)(w1 + kh);
        float2 v11 = *(const float2*)(w1 + 4 + kh);
        float2 v12 = *(const float2*)(w1 + 8);
        b0[0].x = v00.x; b0[0].y = v00.y;
        b0[1].x = v01.x; b0[1].y = v01.y;
        b0[2].x = half ? 0.0f : v02.x; b0[2].y = half ? 0.0f : v02.y;
        b1[0].x = v10.x; b1[0].y = v10.y;
        b1[1].x = v11.x; b1[1].y = v11.y;
        b1[2].x = half ? 0.0f : v12.x; b1[2].y = half ? 0.0f : v12.y;
    }

    // ---- A fragments
    v2f a[3];
    const int p = row0 + mr;
    if (full) {
        const float* arow = in + p * C_IN;
        float2 v0 = *(const float2*)(arow + kh);
        float2 v1 = *(const float2*)(arow + 4 + kh);
        float2 v2 = *(const float2*)(arow + 8);           // k=8 valid for all lanes
        a[0].x = v0.x; a[0].y = v0.y;
        a[1].x = v1.x; a[1].y = v1.y;
        a[2].x = half ? 0.0f : v2.x; a[2].y = half ? 0.0f : v2.y;
    } else {
        const bool pv = (p < N);
#pragma unroll
        for (int ks = 0; ks < 3; ks++) {
            int k = ks * 4 + kh;
            a[ks].x = (pv && k     < C_IN) ? in[p * C_IN + k]     : 0.0f;
            a[ks].y = (pv && k + 1 < C_IN) ? in[p * C_IN + k + 1] : 0.0f;
        }
    }

    v8f c0 = {}, c1 = {};
#pragma unroll
    for (int ks = 0; ks < 3; ks++) {
        c0 = __builtin_amdgcn_wmma_f32_16x16x4_f32(false, a[ks], false, b0[ks],
                                                   (short)0, c0, false, false);
        c1 = __builtin_amdgcn_wmma_f32_16x16x4_f32(false, a[ks], false, b1[ks],
                                                   (short)0, c1, false, false);
    }

    const float s0  = ws[WS_SCALE + mr];
    const float bi0 = ws[WS_BIAS  + mr];
    const float s1  = ws[WS_SCALE + 16 + mr];
    const float bi1 = ws[WS_BIAS  + 16 + mr];
    unsigned int* xmax = (unsigned int*)(ws + WS_XMAX);

    if (full) {
#pragma unroll
        for (int r = 0; r < 8; r++) {
            int pr = row0 + r + 8 * half;                  // all int addressing
            float y0 = fmaxf(c0[r] * s0 + bi0, 0.0f);
            float y1 = fmaxf(c1[r] * s1 + bi1, 0.0f);
            out[pr * 64 + mr]      = y0;
            out[pr * 64 + 16 + mr] = y1;
            int sb = unq[pr] * 32;
            atomicMax(xmax + sb + mr,      __float_as_uint(y0));
            atomicMax(xmax + sb + 16 + mr, __float_as_uint(y1));
        }
    } else {
#pragma unroll
        for (int r = 0; r < 8; r++) {
            int pr = row0 + r + 8 * half;
            if (pr < N) {
                float y0 = fmaxf(c0[r] * s0 + bi0, 0.0f);
                float y1 = fmaxf(c1[r] * s1 + bi1, 0.0f);
                out[pr * 64 + mr]      = y0;
                out[pr * 64 + 16 + mr] = y1;
                int sb = unq[pr] * 32;
                atomicMax(xmax + sb + mr,      __float_as_uint(y0));
                atomicMax(xmax + sb + 16 + mr, __float_as_uint(y1));
            }
        }
    }
}

// ---------------- pass 4: gather per-segment max back to points (float4) ----------------
__global__ void pfn_gather(const int* __restrict__ unq, const float* __restrict__ ws,
                           float* __restrict__ out, int N) {
    int total = N * 8;                                     // N*8 = 16M, int-safe
    int stride = gridDim.x * blockDim.x;
    for (int i = blockIdx.x * blockDim.x + threadIdx.x; i < total; i += stride) {
        int pt = i >> 3;
        int c4 = (i & 7) * 4;
        float4 v = *(const float4*)(ws + WS_XMAX + unq[pt] * 32 + c4);
        *(float4*)(out + pt * 64 + 32 + c4) = v;
    }
}

extern "C" void kernel_launch(void* const* d_in, const int* in_sizes, int n_in,
                              void* d_out, int out_size, void* d_ws, size_t ws_size,
                              hipStream_t stream) {
    const float* in    = (const float*)d_in[0];
    const float* W     = (const float*)d_in[1];
    const float* gamma = (const float*)d_in[2];
    const float* beta  = (const float*)d_in[3];
    const int*   unq   = (const int*)d_in[4];
    const int*   nseg  = (const int*)d_in[5];
    float* out = (float*)d_out;
    float* ws  = (float*)d_ws;

    const int N = in_sizes[0] / C_IN;

    pfn_init_ws<<<4096, 256, 0, stream>>>(ws, nseg);
    pfn_moments<<<1200, 256, 0, stream>>>(in, ws, N);
    pfn_finalize<<<1, 32, 0, stream>>>(W, gamma, beta, ws, 1.0f / (float)N);

    int tiles = (N + 127) / 128;  // 8 waves/block * 16 rows/wave
    pfn_main<<<tiles, 256, 0, stream>>>(in, W, unq, ws, out, N);

    int gblocks = (N * 8 + 255) / 256;
    if (gblocks > 262144) gblocks = 262144;  // grid-stride covers the rest
    pfn_gather<<<gblocks, 256, 0, stream>>>(unq, ws, out, N);
}